// Atom_embedding_MP_87136296501939
// MI455X (gfx1250) — compile-verified
//
#include <hip/hip_runtime.h>

typedef float v2f __attribute__((ext_vector_type(2)));
typedef float v8f __attribute__((ext_vector_type(8)));

#define N_PTS 20000
#define M_PTS 8000
#define DIM   16
#define NLAYER 3
#define HPAD  36     // H=33 padded to multiple of 4
#define EPSV  1e-5f
#define LSLOPE 0.2f

// ---------------------------------------------------------------------------
// Kernel 1: batch-masked KNN, one wave32 per x point.
// ---------------------------------------------------------------------------
__global__ __launch_bounds__(256) void knn_kernel(
    const float* __restrict__ x, const float* __restrict__ y,
    const int* __restrict__ xb, const int* __restrict__ yb,
    int* __restrict__ idx_out, float* __restrict__ dist_out)
{
  const int gtid = blockIdx.x * blockDim.x + threadIdx.x;
  const int wave = gtid >> 5;
  const int lane = threadIdx.x & 31;
  if (wave >= N_PTS) return;

  const float x0 = x[wave * 3 + 0];
  const float x1 = x[wave * 3 + 1];
  const float x2 = x[wave * 3 + 2];
  const int b = xb[wave];

  // y_batch is sorted: binary search segment [lo, hi) of batch b
  int lo, hi;
  {
    int l = 0, r = M_PTS;
    while (l < r) { int mm = (l + r) >> 1; if (yb[mm] < b) l = mm + 1; else r = mm; }
    lo = l;
    r = M_PTS;
    while (l < r) { int mm = (l + r) >> 1; if (yb[mm] <= b) l = mm + 1; else r = mm; }
    hi = l;
  }

  // per-lane sorted (ascending) top-16, all-static indexing
  float dk[16]; int ik[16];
#pragma unroll
  for (int t = 0; t < 16; ++t) { dk[t] = __builtin_inff(); ik[t] = 0; }

  for (int j = lo + lane; j < hi; j += 32) {
    const float d0 = x0 - y[j * 3 + 0];
    const float d1 = x1 - y[j * 3 + 1];
    const float d2 = x2 - y[j * 3 + 2];
    const float v = d0 * d0 + d1 * d1 + d2 * d2;
    if (v < dk[15]) {
#pragma unroll
      for (int t = 15; t >= 1; --t) {
        const bool c = v < dk[t - 1];
        const float nd = c ? dk[t - 1] : v;
        const int   ni = c ? ik[t - 1] : j;
        if (v < dk[t]) { dk[t] = nd; ik[t] = ni; }
      }
      if (v < dk[0]) { dk[0] = v; ik[0] = j; }
    }
  }

  // cross-lane butterfly merge: keep global smallest 16, sorted
#pragma unroll
  for (int msk = 1; msk <= 16; msk <<= 1) {
    float od[16]; int oi[16];
#pragma unroll
    for (int t = 0; t < 16; ++t) {
      od[t] = __shfl_xor(dk[t], msk, 32);
      oi[t] = __shfl_xor(ik[t], msk, 32);
    }
    // bitonic top-k: elementwise min of ascending self vs reversed other
    float md[16]; int mi[16];
#pragma unroll
    for (int t = 0; t < 16; ++t) {
      const bool c = dk[t] <= od[15 - t];
      md[t] = c ? dk[t] : od[15 - t];
      mi[t] = c ? ik[t] : oi[15 - t];
    }
    // bitonic cleanup: strides 8,4,2,1
#pragma unroll
    for (int s = 8; s >= 1; s >>= 1) {
#pragma unroll
      for (int t = 0; t < 16; ++t) {
        if ((t & s) == 0) {
          const int u = t | s;
          if (md[u] < md[t]) {
            const float td = md[t]; md[t] = md[u]; md[u] = td;
            const int   ti = mi[t]; mi[t] = mi[u]; mi[u] = ti;
          }
        }
      }
    }
#pragma unroll
    for (int t = 0; t < 16; ++t) { dk[t] = md[t]; ik[t] = mi[t]; }
  }

  if (lane == 0) {
#pragma unroll
    for (int t = 0; t < 16; ++t) {
      idx_out[wave * 16 + t]  = ik[t];
      dist_out[wave * 16 + t] = dk[t];
    }
  }
}

// ---------------------------------------------------------------------------
// Kernel 2: 3-layer edge-MLP message passing, one wave32 per x point,
// fp32 WMMA (V_WMMA_F32_16X16X4_F32), weights staged in LDS.
// ---------------------------------------------------------------------------
__global__ __launch_bounds__(256) void mp_kernel(
    const float* __restrict__ yat,   // [M,16]
    const int*   __restrict__ idx,   // [N,16]
    const float* __restrict__ dists, // [N,16]
    const float* __restrict__ W1,    // [3,33,33]
    const float* __restrict__ B1,    // [3,33]
    const float* __restrict__ W2,    // [3,16,33]
    const float* __restrict__ B2,    // [3,16]
    const float* __restrict__ GW,    // [3,16]
    const float* __restrict__ GB,    // [3,16]
    float* __restrict__ out)         // [N,16]
{
  __shared__ float sW1[48 * HPAD];   // padded [48][36]
  __shared__ float sW2[16 * HPAD];   // padded [16][36]
  __shared__ float sB1[48];
  __shared__ float sB2[16], sGW[16], sGB[16];
  __shared__ float sH[8][HPAD * 16]; // per-wave h scratch: [h][m]

  const int tid  = threadIdx.x;
  const int wId  = tid >> 5;
  const int lane = tid & 31;
  const int pt   = blockIdx.x * 8 + wId;
  const bool act = (pt < N_PTS);

  const int m   = lane & 15;          // A-matrix row (neighbor index)
  const int o0  = (lane >> 4) << 1;   // K-pair offset: 0 (lanes 0-15) / 2 (16-31)
  const int col = lane & 15;          // B/C/D column within tile

  float pe = 1.0f;                    // point_emb[pt][col] (mirrored in lanes 16-31)
  float dm = 0.0f;
  v2f atf[4];
#pragma unroll
  for (int cc = 0; cc < 4; ++cc) { atf[cc].x = 0.f; atf[cc].y = 0.f; }
  if (act) {
    const int nbr = idx[pt * 16 + m];
    dm = dists[pt * 16 + m];
#pragma unroll
    for (int cc = 0; cc < 4; ++cc) {
      const int j0 = cc * 4 + o0;
      atf[cc].x = yat[nbr * 16 + j0];
      atf[cc].y = yat[nbr * 16 + j0 + 1];
    }
  }

  for (int L = 0; L < NLAYER; ++L) {
    __syncthreads();
    // ---- stage layer weights (zero-padded) into LDS ----
    for (int t = tid; t < 48 * HPAD; t += 256) {
      const int o = t / HPAD, h = t % HPAD;
      sW1[t] = (o < 33 && h < 33) ? W1[(L * 33 + o) * 33 + h] : 0.f;
    }
    for (int t = tid; t < 16 * HPAD; t += 256) {
      const int o = t / HPAD, h = t % HPAD;
      sW2[t] = (h < 33) ? W2[(L * 16 + o) * 33 + h] : 0.f;
    }
    if (tid < 48) sB1[tid] = (tid < 33) ? B1[L * 33 + tid] : 0.f;
    if (tid < 16) {
      sB2[tid] = B2[L * 16 + tid];
      sGW[tid] = GW[L * 16 + tid];
      sGB[tid] = GB[L * 16 + tid];
    }
    __syncthreads();

    if (act) {
      // ---- GEMM1: feats[16x36] @ W1^T -> h[16x48], 3 N-tiles x 9 K-chunks ----
      v8f c0 = {}, c1 = {}, c2 = {};
#pragma unroll
      for (int c = 0; c < 9; ++c) {
        const int k0 = c * 4 + o0;
        v2f a;
        if (c < 4)      { a.x = __shfl(pe, k0, 32); a.y = __shfl(pe, k0 + 1, 32); }
        else if (c < 8) { a = atf[c - 4]; }
        else            { a.x = (o0 == 0) ? dm : 0.f; a.y = 0.f; }
        v2f bA, bB, bC;
        bA.x = sW1[(0  + col) * HPAD + k0]; bA.y = sW1[(0  + col) * HPAD + k0 + 1];
        bB.x = sW1[(16 + col) * HPAD + k0]; bB.y = sW1[(16 + col) * HPAD + k0 + 1];
        bC.x = sW1[(32 + col) * HPAD + k0]; bC.y = sW1[(32 + col) * HPAD + k0 + 1];
        c0 = __builtin_amdgcn_wmma_f32_16x16x4_f32(false, a, false, bA, (short)0, c0, false, false);
        c1 = __builtin_amdgcn_wmma_f32_16x16x4_f32(false, a, false, bB, (short)0, c1, false, false);
        c2 = __builtin_amdgcn_wmma_f32_16x16x4_f32(false, a, false, bC, (short)0, c2, false, false);
      }

      // ---- bias + leakyReLU, write h to per-wave LDS as [h][m] ----
      float* hs = &sH[wId][0];
      const int rbase = (lane >> 4) * 8;  // rows 0-7 / 8-15
#pragma unroll
      for (int t3 = 0; t3 < 3; ++t3) {
        const int n = t3 * 16 + col;
        const v8f cc = (t3 == 0) ? c0 : ((t3 == 1) ? c1 : c2);
        const float bv = (n < 33) ? sB1[n] : 0.f;
        if (n < HPAD) {
#pragma unroll
          for (int r = 0; r < 8; ++r) {
            float v = cc[r] + bv;
            v = (v > 0.f) ? v : v * LSLOPE;
            hs[n * 16 + rbase + r] = v;
          }
        }
      }
      // cross-lane LDS RAW within the wave: wait on split DS counter
      asm volatile("s_wait_dscnt 0x0" ::: "memory");

      // ---- GEMM2: h[16x36] @ W2^T -> msg tiles [16x16] ----
      v8f c3 = {};
#pragma unroll
      for (int c = 0; c < 9; ++c) {
        const int k0 = c * 4 + o0;
        v2f a, bb;
        a.x = hs[k0 * 16 + m];
        a.y = hs[(k0 + 1) * 16 + m];
        bb.x = sW2[col * HPAD + k0];
        bb.y = sW2[col * HPAD + k0 + 1];
        c3 = __builtin_amdgcn_wmma_f32_16x16x4_f32(false, a, false, bb, (short)0, c3, false, false);
      }

      // ---- sum over 16 neighbors (rows) ----
      float part = 0.f;
#pragma unroll
      for (int r = 0; r < 8; ++r) part += c3[r];
      part += __shfl_xor(part, 16, 32);
      const float msg = part + 16.0f * sB2[col];

      // ---- GroupNorm(2,16) over groups of 8 channels via shuffles ----
      float s1 = msg, s2 = msg * msg;
#pragma unroll
      for (int sh = 1; sh <= 4; sh <<= 1) {
        s1 += __shfl_xor(s1, sh, 32);
        s2 += __shfl_xor(s2, sh, 32);
      }
      const float mu  = s1 * 0.125f;
      const float var = s2 * 0.125f - mu * mu;
      float g = (msg - mu) * rsqrtf(var + EPSV) * sGW[col] + sGB[col];
      g = (g > 0.f) ? g : g * LSLOPE;
      pe += g;
    }
  }

  if (act && lane < 16) out[pt * 16 + lane] = pe;
}

// ---------------------------------------------------------------------------
extern "C" void kernel_launch(void* const* d_in, const int* in_sizes, int n_in,
                              void* d_out, int out_size, void* d_ws, size_t ws_size,
                              hipStream_t stream) {
  (void)in_sizes; (void)n_in; (void)out_size; (void)ws_size;
  const float* x   = (const float*)d_in[0];
  const float* y   = (const float*)d_in[1];
  const float* yat = (const float*)d_in[2];
  const int*   xb  = (const int*)d_in[3];
  const int*   yb  = (const int*)d_in[4];
  const float* W1  = (const float*)d_in[5];
  const float* B1  = (const float*)d_in[6];
  const float* W2  = (const float*)d_in[7];
  const float* B2  = (const float*)d_in[8];
  const float* GW  = (const float*)d_in[9];
  const float* GB  = (const float*)d_in[10];
  float* out = (float*)d_out;

  int*   idx = (int*)d_ws;
  float* dst = (float*)((char*)d_ws + (size_t)N_PTS * 16 * sizeof(int));

  knn_kernel<<<(N_PTS + 7) / 8, 256, 0, stream>>>(x, y, xb, yb, idx, dst);
  mp_kernel<<<(N_PTS + 7) / 8, 256, 0, stream>>>(yat, idx, dst, W1, B1, W2, B2, GW, GB, out);
}